// VirtualNodeConvolution_22917945491533
// MI455X (gfx1250) — compile-verified
//
#include <hip/hip_runtime.h>
#include <hip/hip_bf16.h>

// ---------------------------------------------------------------------------
// CDNA5 (gfx1250) wave32 WMMA fp32 implementation of VirtualNodeConvolution.
// ---------------------------------------------------------------------------

#define MUL_S 64
#define MUL_V 32
#define FEAT 160
#define VN 2000
#define NN 200000

#define LXS 164   // padded LDS stride for 160-wide rows (bank-conflict break)
#define LTS 33    // padded LDS stride for 32-wide rows

__device__ __constant__ const float INV8       = 0.125f;                 // 1/sqrt(64)
__device__ __constant__ const float INV_SQRT32 = 0.17677669529663687f;   // 1/sqrt(32)
__device__ __constant__ const float INV_SQRT3  = 0.5773502691896258f;
__device__ __constant__ const float INV_SQRT2  = 0.7071067811865476f;
__device__ __constant__ const float SILU_N     = 1.679059f;
// message scales: 1/sqrt(FAN_TP=96) and segment-mean 1/sqrt(100)
__device__ __constant__ const float SCALE_MS   = 0.010206207261596576f;  // 1/(sqrt(96)*10)
__device__ __constant__ const float SCALE_MV   = 0.005892556509887897f;  // 1/(sqrt(3)*sqrt(96)*10)

typedef __attribute__((ext_vector_type(2))) float v2f;
typedef __attribute__((ext_vector_type(8))) float v8f;

// D = A(16x4) * B(4x16) + C, fp32 WMMA (wave32)
__device__ __forceinline__ v8f wmma4(v2f a, v2f b, v8f c) {
  return __builtin_amdgcn_wmma_f32_16x16x4_f32(
      /*neg_a=*/false, a, /*neg_b=*/false, b,
      /*c_mod=*/(short)0, c, /*reuse_a=*/false, /*reuse_b=*/false);
}

__device__ __forceinline__ float silu_n(float x) {
  return x / (1.0f + __expf(-x)) * SILU_N;
}

// ---------------------------------------------------------------------------
// Kernel 0: zero the segment accumulator (VN x FEAT) in workspace
// ---------------------------------------------------------------------------
__global__ __launch_bounds__(256) void zero_acc_kernel(float* __restrict__ acc) {
  int i = blockIdx.x * 256 + threadIdx.x;
  if (i < VN * FEAT) acc[i] = 0.0f;
}

// ---------------------------------------------------------------------------
// Kernel 1: per-node messages (_fc_tp) + segment scatter (atomics into L2)
// block = 128 threads (4 wave32), 16 node rows per block
// ---------------------------------------------------------------------------
__global__ __launch_bounds__(128) void msg_kernel(
    const float* __restrict__ x_node,   // NN x 160
    const float* __restrict__ sh,       // NN x 4
    const float* __restrict__ WA,       // 64x64
    const float* __restrict__ WB,       // 32x64
    const float* __restrict__ WC,       // 32x32
    const float* __restrict__ WD,       // 64x32
    const int*   __restrict__ batch,    // NN
    float* __restrict__ acc)            // VN x 160
{
  __shared__ float lX[16 * LXS];
  __shared__ float lDot[16 * LTS];
  __shared__ float lT[16 * LTS];
  __shared__ float lSh[16 * 4];
  __shared__ int   lSeg[16];

  const int tid = threadIdx.x;
  const int rowBase = blockIdx.x * 16;

  for (int idx = tid; idx < 16 * FEAT; idx += 128) {
    int r = idx / FEAT, c = idx - r * FEAT;
    lX[r * LXS + c] = x_node[(size_t)rowBase * FEAT + idx];
  }
  for (int idx = tid; idx < 64; idx += 128)
    lSh[idx] = sh[(size_t)rowBase * 4 + idx];
  if (tid < 16) lSeg[tid] = batch[rowBase + tid];
  __syncthreads();

  // dot[r][u] = sum_i v[r][u][i]*sh1[r][i] / sqrt(3)
  for (int idx = tid; idx < 16 * 32; idx += 128) {
    int r = idx >> 5, u = idx & 31;
    const float* vp = &lX[r * LXS + 64 + 3 * u];
    const float* sp = &lSh[r * 4];
    lDot[r * LTS + u] = (vp[0] * sp[1] + vp[1] * sp[2] + vp[2] * sp[3]) * INV_SQRT3;
  }
  __syncthreads();

  const int wave = tid >> 5;
  const int lane = tid & 31;
  const int half = lane >> 4;   // K sub-block select
  const int l    = lane & 15;   // M (A) or N (B,C,D) index

  // ---- m_s tile: wave w -> output columns [w*16, w*16+16) ----
  {
    const int c0 = wave * 16;
    v8f af = {};
    const float s0 = lSh[l * 4 + 0];
    #pragma unroll
    for (int k0 = 0; k0 < 64; k0 += 4) {           // (s*sh0) @ WA
      int kb = k0 + half * 2;
      v2f a; a.x = lX[l * LXS + kb] * s0; a.y = lX[l * LXS + kb + 1] * s0;
      v2f b; b.x = WA[kb * 64 + c0 + l]; b.y = WA[(kb + 1) * 64 + c0 + l];
      af = wmma4(a, b, af);
    }
    #pragma unroll
    for (int k0 = 0; k0 < 32; k0 += 4) {           // dot @ WB
      int kb = k0 + half * 2;
      v2f a; a.x = lDot[l * LTS + kb]; a.y = lDot[l * LTS + kb + 1];
      v2f b; b.x = WB[kb * 64 + c0 + l]; b.y = WB[(kb + 1) * 64 + c0 + l];
      af = wmma4(a, b, af);
    }
    #pragma unroll
    for (int j = 0; j < 8; ++j) {
      int r = j + half * 8;
      atomicAdd(&acc[(size_t)lSeg[r] * FEAT + c0 + l], af[j] * SCALE_MS);
    }
  }

  // ---- T = s @ WD (16x32), waves 0-1 ----
  if (wave < 2) {
    const int c0 = wave * 16;
    v8f tf = {};
    #pragma unroll
    for (int k0 = 0; k0 < 64; k0 += 4) {
      int kb = k0 + half * 2;
      v2f a; a.x = lX[l * LXS + kb]; a.y = lX[l * LXS + kb + 1];
      v2f b; b.x = WD[kb * 32 + c0 + l]; b.y = WD[(kb + 1) * 32 + c0 + l];
      tf = wmma4(a, b, tf);
    }
    #pragma unroll
    for (int j = 0; j < 8; ++j) {
      int r = j + half * 8;
      lT[r * LTS + c0 + l] = tf[j];
    }
  }
  __syncthreads();

  // ---- m_v tiles: t = i*2 + ct  (i = vector component, ct = col tile) ----
  for (int t = wave; t < 6; t += 4) {
    const int i = t >> 1, ct = t & 1, c0 = ct * 16;
    v8f af = {};
    const float s0 = lSh[l * 4 + 0];
    #pragma unroll
    for (int k0 = 0; k0 < 32; k0 += 4) {           // (v_i*sh0) @ WC
      int kb = k0 + half * 2;
      v2f a;
      a.x = lX[l * LXS + 64 + 3 * kb + i] * s0;
      a.y = lX[l * LXS + 64 + 3 * (kb + 1) + i] * s0;
      v2f b; b.x = WC[kb * 32 + c0 + l]; b.y = WC[(kb + 1) * 32 + c0 + l];
      af = wmma4(a, b, af);
    }
    #pragma unroll
    for (int j = 0; j < 8; ++j) {
      int r = j + half * 8;
      float val = (af[j] + lT[r * LTS + c0 + l] * lSh[r * 4 + 1 + i]) * SCALE_MV;
      atomicAdd(&acc[(size_t)lSeg[r] * FEAT + 64 + 3 * (c0 + l) + i], val);
    }
  }
}

// ---------------------------------------------------------------------------
// Kernel 2: x_virtual_out = (o3_linear(x_virtual) + act(acc)) / sqrt(2)
// ---------------------------------------------------------------------------
__global__ __launch_bounds__(128) void vout_kernel(
    const float* __restrict__ xv,     // VN x 160
    const float* __restrict__ Ws,     // 64x64 (Ws_vsc)
    const float* __restrict__ Wv,     // 32x32 (Wv_vsc)
    const float* __restrict__ acc,    // VN x 160 messages
    float* __restrict__ out)          // VN x 160
{
  __shared__ float lX[16 * LXS];
  __shared__ float lM[16 * LXS];

  const int tid = threadIdx.x;
  const int rowBase = blockIdx.x * 16;
  for (int idx = tid; idx < 16 * FEAT; idx += 128) {
    int r = idx / FEAT, c = idx - r * FEAT;
    lX[r * LXS + c] = xv[(size_t)rowBase * FEAT + idx];
    lM[r * LXS + c] = acc[(size_t)rowBase * FEAT + idx];
  }
  __syncthreads();

  const int wave = tid >> 5, lane = tid & 31, half = lane >> 4, l = lane & 15;

  for (int t = wave; t < 10; t += 4) {
    v8f af = {};
    if (t < 4) {                                   // scalar tiles
      const int c0 = t * 16;
      #pragma unroll
      for (int k0 = 0; k0 < 64; k0 += 4) {
        int kb = k0 + half * 2;
        v2f a; a.x = lX[l * LXS + kb]; a.y = lX[l * LXS + kb + 1];
        v2f b; b.x = Ws[kb * 64 + c0 + l]; b.y = Ws[(kb + 1) * 64 + c0 + l];
        af = wmma4(a, b, af);
      }
      #pragma unroll
      for (int j = 0; j < 8; ++j) {
        int r = j + half * 8;
        float m = lM[r * LXS + c0 + l];
        out[(size_t)(rowBase + r) * FEAT + c0 + l] =
            (af[j] * INV8 + silu_n(m)) * INV_SQRT2;
      }
    } else {                                       // vector tiles
      const int i = (t - 4) >> 1, ct = (t - 4) & 1, c0 = ct * 16;
      #pragma unroll
      for (int k0 = 0; k0 < 32; k0 += 4) {
        int kb = k0 + half * 2;
        v2f a;
        a.x = lX[l * LXS + 64 + 3 * kb + i];
        a.y = lX[l * LXS + 64 + 3 * (kb + 1) + i];
        v2f b; b.x = Wv[kb * 32 + c0 + l]; b.y = Wv[(kb + 1) * 32 + c0 + l];
        af = wmma4(a, b, af);
      }
      #pragma unroll
      for (int j = 0; j < 8; ++j) {
        int r = j + half * 8;
        int col = 64 + 3 * (c0 + l) + i;
        float m = lM[r * LXS + col];
        out[(size_t)(rowBase + r) * FEAT + col] =
            (af[j] * INV_SQRT32 + m) * INV_SQRT2;
      }
    }
  }
}

// ---------------------------------------------------------------------------
// Kernel 3: x_node_out = (o3_linear(x_node) + act(o3_linear(xv_out[batch])))/sqrt2
// ---------------------------------------------------------------------------
__global__ __launch_bounds__(128) void nout_kernel(
    const float* __restrict__ xn,      // NN x 160
    const float* __restrict__ Ws_nsc,  // 64x64
    const float* __restrict__ Wv_nsc,  // 32x32
    const float* __restrict__ Ws_n2v,  // 64x64
    const float* __restrict__ Wv_n2v,  // 32x32
    const int*   __restrict__ batch,   // NN
    const float* __restrict__ xv_out,  // VN x 160 (virtual output, L2-resident)
    float* __restrict__ out)           // NN x 160
{
  __shared__ float lX[16 * LXS];
  __shared__ float lG[16 * LXS];
  __shared__ int   lSeg[16];

  const int tid = threadIdx.x;
  const int rowBase = blockIdx.x * 16;

  if (tid < 16) lSeg[tid] = batch[rowBase + tid];
  for (int idx = tid; idx < 16 * FEAT; idx += 128) {
    int r = idx / FEAT, c = idx - r * FEAT;
    lX[r * LXS + c] = xn[(size_t)rowBase * FEAT + idx];
  }
  __syncthreads();
  for (int idx = tid; idx < 16 * FEAT; idx += 128) {
    int r = idx / FEAT, c = idx - r * FEAT;
    lG[r * LXS + c] = xv_out[(size_t)lSeg[r] * FEAT + c];  // gather (L2 hit)
  }
  __syncthreads();

  const int wave = tid >> 5, lane = tid & 31, half = lane >> 4, l = lane & 15;

  for (int t = wave; t < 10; t += 4) {
    v8f sn = {}, gn = {};
    if (t < 4) {                                   // scalar tiles
      const int c0 = t * 16;
      #pragma unroll
      for (int k0 = 0; k0 < 64; k0 += 4) {
        int kb = k0 + half * 2;
        v2f a; a.x = lX[l * LXS + kb]; a.y = lX[l * LXS + kb + 1];
        v2f b; b.x = Ws_nsc[kb * 64 + c0 + l]; b.y = Ws_nsc[(kb + 1) * 64 + c0 + l];
        sn = wmma4(a, b, sn);
        v2f ag; ag.x = lG[l * LXS + kb]; ag.y = lG[l * LXS + kb + 1];
        v2f bg; bg.x = Ws_n2v[kb * 64 + c0 + l]; bg.y = Ws_n2v[(kb + 1) * 64 + c0 + l];
        gn = wmma4(ag, bg, gn);
      }
      #pragma unroll
      for (int j = 0; j < 8; ++j) {
        int r = j + half * 8;
        out[(size_t)(rowBase + r) * FEAT + c0 + l] =
            (sn[j] * INV8 + silu_n(gn[j] * INV8)) * INV_SQRT2;
      }
    } else {                                       // vector tiles
      const int i = (t - 4) >> 1, ct = (t - 4) & 1, c0 = ct * 16;
      #pragma unroll
      for (int k0 = 0; k0 < 32; k0 += 4) {
        int kb = k0 + half * 2;
        v2f a;
        a.x = lX[l * LXS + 64 + 3 * kb + i];
        a.y = lX[l * LXS + 64 + 3 * (kb + 1) + i];
        v2f b; b.x = Wv_nsc[kb * 32 + c0 + l]; b.y = Wv_nsc[(kb + 1) * 32 + c0 + l];
        sn = wmma4(a, b, sn);
        v2f ag;
        ag.x = lG[l * LXS + 64 + 3 * kb + i];
        ag.y = lG[l * LXS + 64 + 3 * (kb + 1) + i];
        v2f bg; bg.x = Wv_n2v[kb * 32 + c0 + l]; bg.y = Wv_n2v[(kb + 1) * 32 + c0 + l];
        gn = wmma4(ag, bg, gn);
      }
      #pragma unroll
      for (int j = 0; j < 8; ++j) {
        int r = j + half * 8;
        int col = 64 + 3 * (c0 + l) + i;
        out[(size_t)(rowBase + r) * FEAT + col] =
            (sn[j] * INV_SQRT32 + gn[j] * INV_SQRT32) * INV_SQRT2;
      }
    }
  }
}

// ---------------------------------------------------------------------------
// Host launcher
// ---------------------------------------------------------------------------
extern "C" void kernel_launch(void* const* d_in, const int* in_sizes, int n_in,
                              void* d_out, int out_size, void* d_ws, size_t ws_size,
                              hipStream_t stream) {
  const float* x_virtual = (const float*)d_in[0];
  const float* x_node    = (const float*)d_in[1];
  const float* sh        = (const float*)d_in[2];
  const float* Ws_vsc    = (const float*)d_in[3];
  const float* Wv_vsc    = (const float*)d_in[4];
  const float* Ws_nsc    = (const float*)d_in[5];
  const float* Wv_nsc    = (const float*)d_in[6];
  const float* WA        = (const float*)d_in[7];
  const float* WB        = (const float*)d_in[8];
  const float* WC        = (const float*)d_in[9];
  const float* WD        = (const float*)d_in[10];
  const float* Ws_n2v    = (const float*)d_in[11];
  const float* Wv_n2v    = (const float*)d_in[12];
  const int*   batch     = (const int*)d_in[13];

  float* acc    = (float*)d_ws;                 // VN x FEAT segment accumulator
  float* out_v  = (float*)d_out;                // x_virtual_out
  float* out_n  = out_v + (size_t)VN * FEAT;    // x_node_out

  zero_acc_kernel<<<dim3((VN * FEAT + 255) / 256), dim3(256), 0, stream>>>(acc);
  msg_kernel<<<dim3(NN / 16), dim3(128), 0, stream>>>(
      x_node, sh, WA, WB, WC, WD, batch, acc);
  vout_kernel<<<dim3(VN / 16), dim3(128), 0, stream>>>(
      x_virtual, Ws_vsc, Wv_vsc, acc, out_v);
  nout_kernel<<<dim3(NN / 16), dim3(128), 0, stream>>>(
      x_node, Ws_nsc, Wv_nsc, Ws_n2v, Wv_n2v, batch, out_v, out_n);
}